// GLRK2nd_33758442947048
// MI455X (gfx1250) — compile-verified
//
#include <hip/hip_runtime.h>

typedef __attribute__((ext_vector_type(2))) float v2f;
typedef __attribute__((ext_vector_type(4))) float f4;
typedef __attribute__((ext_vector_type(8))) float v8f;

#define DM 64
#define LSTR 68   // LDS row stride (pad 64->68: conflict-free WMMA operand gathers)

__global__ __launch_bounds__(128) void glrk2_kernel(
    const float* __restrict__ A1, const float* __restrict__ y0,
    const float* __restrict__ hptr, float* __restrict__ out_y,
    float* __restrict__ out_A)
{
    __shared__ float sA [DM * LSTR];
    __shared__ float sA2[DM * LSTR];
    __shared__ float vb0[DM];
    __shared__ float vb1[DM];

    const int tid  = threadIdx.x;
    const int b    = blockIdx.x;
    const int lane = tid & 31;
    const int wave = tid >> 5;      // 0..3
    const int half = lane >> 4;     // 0/1
    const int l16  = lane & 15;

    const float hs = hptr[0];
    const float e1 = 0.5f * hs;     // A_COEF * h
    const float e2 = e1 * e1;
    const float e4 = e2 * e2;

    const float* Ab   = A1    + (size_t)b * (DM * DM);
    float*       Aout = out_A + (size_t)b * (DM * DM);

    // ---- Stream A: global -> LDS, and the verbatim copy to output (NT path) ----
    #pragma unroll
    for (int it = 0; it < 8; ++it) {
        const int t   = tid + it * 128;        // float4 index 0..1023
        const int row = t >> 4;                // 64 floats per row = 16 float4
        const int col = (t & 15) << 2;
        const f4 v = __builtin_nontemporal_load((const f4*)Ab + t);
        __builtin_nontemporal_store(v, (f4*)Aout + t);
        float* dst = &sA[row * LSTR + col];
        dst[0] = v.x; dst[1] = v.y; dst[2] = v.z; dst[3] = v.w;
    }
    __syncthreads();

    // ---- A2 = A * A via V_WMMA_F32_16X16X4_F32 ----
    // Wave w owns output rows [16w, 16w+16). A-operand row for this lane:
    const int arow = wave * 16 + l16;
    // Cache A-operand for all 16 K-steps (ISA: VGPR0 <-> K=k0+2*half, VGPR1 <-> +1)
    float a0c[16], a1c[16];
    #pragma unroll
    for (int kk = 0; kk < 16; ++kk) {
        const int k = kk * 4 + half * 2;
        a0c[kk] = sA[arow * LSTR + k];
        a1c[kk] = sA[arow * LSTR + k + 1];
    }
    #pragma unroll
    for (int c = 0; c < 4; ++c) {
        v8f acc = {};
        #pragma unroll
        for (int kk = 0; kk < 16; ++kk) {
            const int k = kk * 4 + half * 2;
            v2f a; a.x = a0c[kk]; a.y = a1c[kk];
            v2f bb;                              // B 4x16: N=l16, K=k0+2*half+vgpr
            bb.x = sA[k * LSTR + c * 16 + l16];
            bb.y = sA[(k + 1) * LSTR + c * 16 + l16];
            acc = __builtin_amdgcn_wmma_f32_16x16x4_f32(
                false, a, false, bb, (short)0, acc, false, false);
        }
        // D layout: VGPR r -> M = r + 8*half, N = l16
        #pragma unroll
        for (int r = 0; r < 8; ++r)
            sA2[(wave * 16 + r + half * 8) * LSTR + c * 16 + l16] = acc[r];
    }
    __syncthreads();

    // ---- z = (I+eA)(I+e2*A2)(I+e4*A2*A2) y0 ; k1 = A z ; y = y0 + h*k1 ----
    float y0i = 0.0f, z = 0.0f, s;
    if (tid < DM) { y0i = y0[(size_t)b * DM + tid]; vb0[tid] = y0i; }
    __syncthreads();

    // t = A2*y0 -> vb1
    if (tid < DM) {
        s = 0.0f;
        #pragma unroll 8
        for (int j = 0; j < DM; ++j) s += sA2[tid * LSTR + j] * vb0[j];
        vb1[tid] = s;
    }
    __syncthreads();

    // z1 = y0 + e4*A2*t -> vb0
    if (tid < DM) {
        s = 0.0f;
        #pragma unroll 8
        for (int j = 0; j < DM; ++j) s += sA2[tid * LSTR + j] * vb1[j];
        z = y0i + e4 * s; vb0[tid] = z;
    }
    __syncthreads();

    // z2 = z1 + e2*A2*z1 -> vb1
    if (tid < DM) {
        s = 0.0f;
        #pragma unroll 8
        for (int j = 0; j < DM; ++j) s += sA2[tid * LSTR + j] * vb0[j];
        z = z + e2 * s; vb1[tid] = z;
    }
    __syncthreads();

    // z3 = z2 + e1*A*z2 -> vb0
    if (tid < DM) {
        s = 0.0f;
        #pragma unroll 8
        for (int j = 0; j < DM; ++j) s += sA[tid * LSTR + j] * vb1[j];
        z = z + e1 * s; vb0[tid] = z;
    }
    __syncthreads();

    // k1 = A*z3 ; y_next = y0 + h*k1   (B_COEF = 1)
    if (tid < DM) {
        s = 0.0f;
        #pragma unroll 8
        for (int j = 0; j < DM; ++j) s += sA[tid * LSTR + j] * vb0[j];
        out_y[(size_t)b * DM + tid] = y0i + hs * s;
    }
}

extern "C" void kernel_launch(void* const* d_in, const int* in_sizes, int n_in,
                              void* d_out, int out_size, void* d_ws, size_t ws_size,
                              hipStream_t stream) {
    const float* A1 = (const float*)d_in[0];
    const float* y0 = (const float*)d_in[1];
    const float* h  = (const float*)d_in[2];
    const int B = in_sizes[1] / DM;           // 16384
    float* out_y = (float*)d_out;
    float* out_A = out_y + (size_t)B * DM;    // tuple order: (y_next, A1)
    hipLaunchKernelGGL(glrk2_kernel, dim3(B), dim3(128), 0, stream,
                       A1, y0, h, out_y, out_A);
}